// RecursiveNonlinearEquationSolver_67095979098265
// MI455X (gfx1250) — compile-verified
//
#include <hip/hip_runtime.h>

// Implicit midpoint ODE solver, Newton + block-bidiagonal Thomas sweep.
// Inner 64x64 solves replaced by a 6-step Neumann fixed point (D = I + O(DT)),
// so all matrix work becomes batched 64-wide matvecs against a shared W,
// executed with V_WMMA_F32_16X16X4_F32 (full fp32, matches reference dtype).

#define S_DIM   64
#define C_DIM   8
#define N_ITER  4
#define NEUMANN 6          // even -> final iterate lands in sX0
#define B_TOT   128
#define NB      8          // (65-1)/C_DIM
#define BSUB    8          // batch rows per workgroup (independent slices)
#define NWG     (B_TOT / BSUB)
#define THREADS 128        // 4 waves (wave32)
#define DT_C    0.05f
#define LDA     66         // padded stride for WMMA A/B staging (bank spread)

typedef __attribute__((ext_vector_type(2))) float v2f;
typedef __attribute__((ext_vector_type(8))) float v8f;

// One 16x16 output tile of (16 x 64) @ (64 x 64), K accumulated in steps of 4.
// A: rows 0..15 at stride LDA (rows 8..15 pre-zeroed). Bt: Wt[k][n] at stride LDA.
__device__ __forceinline__ v8f wmma_gemm_16x64(const float* __restrict__ A,
                                               const float* __restrict__ Bt,
                                               int n0, int lane) {
  v8f acc = {0.f, 0.f, 0.f, 0.f, 0.f, 0.f, 0.f, 0.f};
  const int m = lane & 15;
  const int h = (lane >> 4) << 1;  // K sub-offset: 0 (lanes 0-15) or 2 (lanes 16-31)
#pragma unroll
  for (int k0 = 0; k0 < S_DIM; k0 += 4) {
    v2f a, b;
    a.x = A[m * LDA + k0 + h];
    a.y = A[m * LDA + k0 + h + 1];
    b.x = Bt[(k0 + h) * LDA + n0 + m];
    b.y = Bt[(k0 + h + 1) * LDA + n0 + m];
    acc = __builtin_amdgcn_wmma_f32_16x16x4_f32(false, a, false, b,
                                                (short)0, acc, false, false);
  }
  return acc;
}

__global__ __launch_bounds__(THREADS, 1)
void solver_gfx1250_kernel(const float* __restrict__ y0,
                           const float* __restrict__ forces,
                           const float* __restrict__ W,
                           float* __restrict__ out) {
  __shared__ float sWt[S_DIM * LDA];       // W transposed: Wt[s][p] = W[p][s]
  __shared__ float sS [16 * LDA];          // GEMM A staging (rows 8..15 zero)
  __shared__ float sX0[16 * LDA];          // Neumann ping  (rows 8..15 zero)
  __shared__ float sX1[16 * LDA];          // Neumann pong  (rows 8..15 zero)
  __shared__ float sY[C_DIM][BSUB * S_DIM];
  __shared__ float sYprev[BSUB * S_DIM];
  __shared__ float sYm  [BSUB * S_DIM];    // old y_{c-1} during sweep
  __shared__ float sT   [BSUB * S_DIM];
  __shared__ float sG   [BSUB * S_DIM];
  __shared__ float sRhs [BSUB * S_DIM];
  __shared__ float sWx  [BSUB * S_DIM];    // W @ x_prev

  const int tid  = threadIdx.x;
  const int lane = tid & 31;
  const int wave = tid >> 5;
  const int n0   = wave * 16;              // this wave's 16-column tile
  const int b0   = blockIdx.x * BSUB;      // batch slice base

  // --- init: Wt, zero staging buffers, y_prev = y0, emit out step 0 ---
  for (int e = tid; e < S_DIM * S_DIM; e += THREADS) {
    int p = e >> 6, s = e & 63;
    sWt[s * LDA + p] = W[p * S_DIM + s];
  }
  for (int e = tid; e < 16 * LDA; e += THREADS) {
    sS[e] = 0.f; sX0[e] = 0.f; sX1[e] = 0.f;
  }
  for (int e = tid; e < BSUB * S_DIM; e += THREADS) {
    int b = e >> 6, j = e & 63;
    float v = y0[(b0 + b) * S_DIM + j];
    sYprev[e] = v;
    out[(size_t)(b0 + b) * S_DIM + j] = v;
  }
  __syncthreads();

  for (int blk = 0; blk < NB; ++blk) {
    const float* fblk =
        forces + ((size_t)(1 + blk * C_DIM) * B_TOT + b0) * S_DIM;
    // prefetch this block's force slab (gfx1250 global_prefetch_b8)
    for (int off = tid * 16; off < C_DIM * BSUB * S_DIM; off += THREADS * 16)
      __builtin_prefetch(fblk + off, 0, 1);

    // y chunk starts at zero each block (matches reference)
    for (int c = 0; c < C_DIM; ++c)
      for (int e = tid; e < BSUB * S_DIM; e += THREADS) sY[c][e] = 0.f;
    __syncthreads();

    for (int it = 0; it < N_ITER; ++it) {
      for (int e = tid; e < BSUB * S_DIM; e += THREADS) sYm[e] = sYprev[e];
      __syncthreads();

      for (int c = 0; c < C_DIM; ++c) {
        // s = 0.5*(y_{c-1,old} + y_{c,old})
        for (int e = tid; e < BSUB * S_DIM; e += THREADS) {
          int b = e >> 6, j = e & 63;
          sS[b * LDA + j] = 0.5f * (sYm[e] + sY[c][e]);
        }
        __syncthreads();

        // GEMM1: t = tanh(s @ Wt); GEMM2 (c>0): Wx = x_prev @ Wt
        {
          v8f tacc = wmma_gemm_16x64(sS, sWt, n0, lane);
          v8f wacc = {0.f, 0.f, 0.f, 0.f, 0.f, 0.f, 0.f, 0.f};
          if (c > 0) wacc = wmma_gemm_16x64(sX0, sWt, n0, lane);
          if ((lane >> 4) == 0) {           // rows 0..7 (valid batch rows)
            int col = n0 + (lane & 15);
#pragma unroll
            for (int v = 0; v < 8; ++v) {
              sT[v * S_DIM + col] = tanhf(tacc[v]);
              if (c > 0) sWx[v * S_DIM + col] = wacc[v];
            }
          }
        }
        __syncthreads();

        // residual, Jacobian diag scale g, rhs = R - L x_prev; seed x0 = rhs
        for (int e = tid; e < BSUB * S_DIM; e += THREADS) {
          int b = e >> 6, j = e & 63;
          float t = sT[e];
          float g = 0.5f * DT_C * (1.0f - t * t);
          float fv = fblk[(size_t)c * B_TOT * S_DIM + b * S_DIM + j];
          float R = sY[c][e] - sYm[e] + DT_C * t - DT_C * fv;
          float rhs = R;
          if (c > 0) {
            float xp = sX0[b * LDA + j];          // L x = -x + g*(W x)
            rhs += xp - g * sWx[e];
          }
          sG[e] = g;
          sRhs[e] = rhs;
          sX0[b * LDA + j] = rhs;                 // Neumann x0
        }
        __syncthreads();

        // Neumann solve of (I + diag(g) W) x = rhs :  x <- rhs - g*(W x)
        for (int k = 0; k < NEUMANN; ++k) {
          const float* xin  = (k & 1) ? sX1 : sX0;
          float*       xout = (k & 1) ? sX0 : sX1;
          v8f macc = wmma_gemm_16x64(xin, sWt, n0, lane);
          if ((lane >> 4) == 0) {
            int col = n0 + (lane & 15);
#pragma unroll
            for (int v = 0; v < 8; ++v) {
              int e = v * S_DIM + col;
              xout[v * LDA + col] = sRhs[e] - sG[e] * macc[v];
            }
          }
          __syncthreads();
        }
        // NEUMANN even -> solution in sX0 (also serves as x_prev next step)

        // save old y_c for next step's ym, apply Newton update y_c -= x_c
        for (int e = tid; e < BSUB * S_DIM; e += THREADS) {
          int b = e >> 6, j = e & 63;
          float yold = sY[c][e];
          sYm[e] = yold;
          sY[c][e] = yold - sX0[b * LDA + j];
        }
        __syncthreads();
      }  // c
    }    // Newton it

    // emit chunk, carry y_prev
    for (int c = 0; c < C_DIM; ++c)
      for (int e = tid; e < BSUB * S_DIM; e += THREADS) {
        int b = e >> 6, j = e & 63;
        out[((size_t)(1 + blk * C_DIM + c) * B_TOT + b0 + b) * S_DIM + j] =
            sY[c][e];
      }
    for (int e = tid; e < BSUB * S_DIM; e += THREADS)
      sYprev[e] = sY[C_DIM - 1][e];
    __syncthreads();
  }  // blk
}

extern "C" void kernel_launch(void* const* d_in, const int* in_sizes, int n_in,
                              void* d_out, int out_size, void* d_ws,
                              size_t ws_size, hipStream_t stream) {
  (void)in_sizes; (void)n_in; (void)out_size; (void)d_ws; (void)ws_size;
  const float* y0     = (const float*)d_in[0];
  const float* forces = (const float*)d_in[1];
  const float* W      = (const float*)d_in[2];
  // d_in[3] is n (==65), baked into NB
  float* out = (float*)d_out;
  solver_gfx1250_kernel<<<NWG, THREADS, 0, stream>>>(y0, forces, W, out);
}